// RecurrentGCN_28896539967830
// MI455X (gfx1250) — compile-verified
//
#include <hip/hip_runtime.h>
#include <hip/hip_bf16.h>
#include <math.h>

typedef __attribute__((ext_vector_type(2))) float v2f;
typedef __attribute__((ext_vector_type(8))) float v8f;

#define WAVES_PER_BLOCK 8

__device__ __forceinline__ void atomAddF(float* p, float v) {
    (void)__hip_atomic_fetch_add(p, v, __ATOMIC_RELAXED, __HIP_MEMORY_SCOPE_AGENT);
}

__device__ __forceinline__ float sigm(float x) {
    return 1.0f / (1.0f + __expf(-x));
}

// ---------------------------------------------------------------------------
// Swizzle W_big = [Wx;Rx;Wh;Rh] (160 x 64 f32) into per-fragment layout:
//   bsw[((k*4+g)*32 + lane)*2 + j] = W_big[4k + 2*(lane>=16) + j][g*16 + (lane&15)]
// so each wave's B-fragment load is one coalesced b64 per lane.
// ---------------------------------------------------------------------------
__global__ void lrgcn_swizzle_w(const float* __restrict__ Wx, const float* __restrict__ Rx,
                                const float* __restrict__ Wh, const float* __restrict__ Rh,
                                float* __restrict__ bsw) {
    int t = blockIdx.x * blockDim.x + threadIdx.x;
    if (t >= 40 * 4 * 32 * 2) return;
    int j = t & 1;
    int l = (t >> 1) & 31;
    int g = (t >> 6) & 3;
    int k = t >> 8;                     // 0..39
    int row = 4 * k + 2 * (l >> 4) + j; // 0..159
    int col = l & 15;
    float v;
    if (row < 64)       v = Wx[(g * 64 + row) * 16 + col];
    else if (row < 128) v = Rx[(g * 64 + (row - 64)) * 16 + col];
    else if (row < 144) v = Wh[(g * 16 + (row - 128)) * 16 + col];
    else                v = Rh[(g * 16 + (row - 144)) * 16 + col];
    bsw[t] = v;
}

// ---------------------------------------------------------------------------
// Edge scatter: thread = (edge, chunk). chunks 0..15 -> x (64 f32),
// chunks 16..19 -> h0 (16 f32). Fire-and-forget f32 atomics into L2.
// ---------------------------------------------------------------------------
__global__ void lrgcn_scatter(const float* __restrict__ x, const float* __restrict__ h0,
                              const long long* __restrict__ ei,
                              float* __restrict__ aggx, float* __restrict__ aggh,
                              float* __restrict__ deg, int E) {
    long long tid = (long long)blockIdx.x * blockDim.x + threadIdx.x;
    long long total = (long long)E * 20;
    if (tid >= total) return;
    int e = (int)(tid / 20);
    int c = (int)(tid - (long long)e * 20);
    int src = (int)ei[e];
    int dst = (int)ei[(long long)E + e];
    if (c < 16) {
        const float4 v = *(const float4*)(x + (size_t)src * 64 + c * 4);
        float* p = aggx + (size_t)dst * 64 + c * 4;
        atomAddF(p + 0, v.x); atomAddF(p + 1, v.y);
        atomAddF(p + 2, v.z); atomAddF(p + 3, v.w);
        if (c == 0) atomAddF(deg + dst, 1.0f);
    } else {
        int cc = c - 16;
        const float4 v = *(const float4*)(h0 + (size_t)src * 16 + cc * 4);
        float* p = aggh + (size_t)dst * 16 + cc * 4;
        atomAddF(p + 0, v.x); atomAddF(p + 1, v.y);
        atomAddF(p + 2, v.z); atomAddF(p + 3, v.w);
    }
}

// ---------------------------------------------------------------------------
// Fused GEMM (K=160, fp32 WMMA 16x16x4) + LSTM gate epilogue.
// One wave handles 16 nodes and ALL 4 gate tiles (4 v8f accumulators),
// so the gate mix is per-lane VALU with no LDS exchange.
// A row = [agg_x*invdeg (64) | x (64) | agg_h*invdeg (16) | h0 (16)].
// ---------------------------------------------------------------------------
__global__ void __launch_bounds__(32 * WAVES_PER_BLOCK)
lrgcn_gemm(const float* __restrict__ x, const float* __restrict__ h0,
           const float* __restrict__ c0, const float* __restrict__ aggx,
           const float* __restrict__ aggh, const float* __restrict__ deg,
           const float* __restrict__ bsw, const float* __restrict__ bx,
           const float* __restrict__ bh, const float* __restrict__ lin_w,
           const float* __restrict__ lin_b, float* __restrict__ out,
           float* __restrict__ h_out, float* __restrict__ c_out,
           int numTiles, int N) {
    const int lane = threadIdx.x & 31;
    const int wave = threadIdx.x >> 5;
    const int tile = blockIdx.x * WAVES_PER_BLOCK + wave;
    if (tile >= numTiles) return;   // whole-wave uniform exit; EXEC stays full

    const int half = lane >> 4;     // A/B VGPR K-pair selector
    int rowNode = tile * 16 + (lane & 15);
    if (rowNode >= N) rowNode = N - 1;   // clamp loads for partial tile

    const float d = deg[rowNode];
    const float invd = 1.0f / fmaxf(d, 1.0f);

    const float* Ax = aggx + (size_t)rowNode * 64 + 2 * half;
    const float* Xx = x    + (size_t)rowNode * 64 + 2 * half;
    const float* Ah = aggh + (size_t)rowNode * 16 + 2 * half;
    const float* Hh = h0   + (size_t)rowNode * 16 + 2 * half;

    v8f acc0 = {}, acc1 = {}, acc2 = {}, acc3 = {};

#pragma unroll
    for (int k = 0; k < 40; ++k) {
        const float* ap;
        float s;
        if (k < 16)      { ap = Ax + 4 * k;        s = invd; }
        else if (k < 32) { ap = Xx + 4 * k - 64;   s = 1.0f; }
        else if (k < 36) { ap = Ah + 4 * k - 128;  s = invd; }
        else             { ap = Hh + 4 * k - 144;  s = 1.0f; }
        v2f a;
        a[0] = ap[0] * s;
        a[1] = ap[1] * s;
        const float* bp = bsw + k * 256 + lane * 2;   // (k*4+g)*32*2 + lane*2
        const v2f b0 = *(const v2f*)(bp);
        const v2f b1 = *(const v2f*)(bp + 64);
        const v2f b2 = *(const v2f*)(bp + 128);
        const v2f b3 = *(const v2f*)(bp + 192);
        acc0 = __builtin_amdgcn_wmma_f32_16x16x4_f32(false, a, false, b0, (short)0, acc0, false, false);
        acc1 = __builtin_amdgcn_wmma_f32_16x16x4_f32(false, a, false, b1, (short)0, acc1, false, false);
        acc2 = __builtin_amdgcn_wmma_f32_16x16x4_f32(false, a, false, b2, (short)0, acc2, false, false);
        acc3 = __builtin_amdgcn_wmma_f32_16x16x4_f32(false, a, false, b3, (short)0, acc3, false, false);
    }

    // Epilogue: C/D layout: vgpr r, lanes0-15 -> M=r, lanes16-31 -> M=r+8, N=lane&15
    const int hid = lane & 15;
    const float bI = bx[hid]      + bh[hid];
    const float bF = bx[16 + hid] + bh[16 + hid];
    const float bC = bx[32 + hid] + bh[32 + hid];
    const float bO = bx[48 + hid] + bh[48 + hid];
    const float lw = lin_w[hid];
    const float lb = lin_b[0];

#pragma unroll
    for (int r = 0; r < 8; ++r) {
        const int m = tile * 16 + r + half * 8;
        const bool ok = (m < N);
        const float c0v = ok ? c0[(size_t)m * 16 + hid] : 0.0f;
        const float iG = sigm(acc0[r] + bI);
        const float fG = sigm(acc1[r] + bF);
        const float tG = tanhf(acc2[r] + bC);
        const float oG = sigm(acc3[r] + bO);
        const float cn = fG * c0v + iG * tG;
        const float hn = oG * tanhf(cn);
        if (ok) {
            h_out[(size_t)m * 16 + hid] = hn;
            c_out[(size_t)m * 16 + hid] = cn;
        }
        float v = fmaxf(hn, 0.0f) * lw;
        v += __shfl_xor(v, 1, 32);
        v += __shfl_xor(v, 2, 32);
        v += __shfl_xor(v, 4, 32);
        v += __shfl_xor(v, 8, 32);   // sums within each 16-lane half
        if (hid == 0 && ok) out[m] = v + lb;
    }
}

extern "C" void kernel_launch(void* const* d_in, const int* in_sizes, int n_in,
                              void* d_out, int out_size, void* d_ws, size_t ws_size,
                              hipStream_t stream) {
    const float*     x   = (const float*)d_in[0];
    const long long* ei  = (const long long*)d_in[1];
    // d_in[2] = edge_weight (all relation 0) -> drops out of the math
    const float*     h0  = (const float*)d_in[3];
    const float*     c0  = (const float*)d_in[4];
    const float*     Wx  = (const float*)d_in[5];
    const float*     Rx  = (const float*)d_in[6];
    const float*     bx  = (const float*)d_in[7];
    const float*     Wh  = (const float*)d_in[8];
    const float*     Rh  = (const float*)d_in[9];
    const float*     bh  = (const float*)d_in[10];
    const float*     lw  = (const float*)d_in[11];
    const float*     lb  = (const float*)d_in[12];

    const int N = in_sizes[0] / 64;
    const int E = in_sizes[1] / 2;

    float* ws   = (float*)d_ws;
    float* aggx = ws;                               // N*64
    float* aggh = aggx + (size_t)N * 64;            // N*16
    float* deg  = aggh + (size_t)N * 16;            // N
    float* bsw  = deg + N;                          // 10240

    // zero the scatter accumulators + degree each call (deterministic)
    hipMemsetAsync(ws, 0, sizeof(float) * ((size_t)N * 81), stream);

    lrgcn_swizzle_w<<<(10240 + 255) / 256, 256, 0, stream>>>(Wx, Rx, Wh, Rh, bsw);

    const long long totalScatter = (long long)E * 20;
    const int scatterBlocks = (int)((totalScatter + 255) / 256);
    lrgcn_scatter<<<scatterBlocks, 256, 0, stream>>>(x, h0, ei, aggx, aggh, deg, E);

    const int numTiles = (N + 15) / 16;
    const int gemmBlocks = (numTiles + WAVES_PER_BLOCK - 1) / WAVES_PER_BLOCK;
    float* out  = (float*)d_out;
    float* hout = out + N;
    float* cout = hout + (size_t)N * 16;
    lrgcn_gemm<<<gemmBlocks, 32 * WAVES_PER_BLOCK, 0, stream>>>(
        x, h0, c0, aggx, aggh, deg, bsw, bx, bh, lw, lb, out, hout, cout, numTiles, N);
}